// OneBodySphericalToInterleaved_7043746365489
// MI455X (gfx1250) — compile-verified
//
#include <hip/hip_runtime.h>
#include <hip/hip_bf16.h>
#include <stdint.h>

// ---------------------------------------------------------------------------
// Ragged gather/compaction:
//   out = concat_i( src[i, map[e_i][0..K_{e_i})] )
// Memory-bound (~173 MB => ~7.5us @ 23.3 TB/s). CDNA5 path: async
// global->LDS row staging (global_load_async_to_lds_b128, ASYNCcnt),
// LDS permutation gather, coalesced NT stores.
// ---------------------------------------------------------------------------

#define PAD        128
#define APB        256   // atoms per block (gather/count kernels)
#define GATHER_TPB 256   // 8 waves of 32
#define SCAN_TPB   1024

// ---- CDNA5 async global->LDS helpers (inline asm; portable across toolchains)

__device__ __forceinline__ void async_row_b128(const float* gaddr, uint32_t lds_byte) {
    // Each lane copies 16B from the global address gaddr to the LDS byte address lds_byte  (tracked by ASYNCcnt)
    uint64_t ga = (uint64_t)(uintptr_t)gaddr;
    asm volatile("global_load_async_to_lds_b128 %0, %1, off"
                 :: "v"(lds_byte), "v"(ga)
                 : "memory");
}

__device__ __forceinline__ void wait_async_le0() {
    asm volatile("s_wait_asynccnt 0" ::: "memory");
}
__device__ __forceinline__ void wait_async_le1() {
    asm volatile("s_wait_asynccnt 1" ::: "memory");
}

// ---- Pass 0: per-element valid count (mask is a prefix mask; sum == K_e) ----

__global__ void elem_counts_kernel(const int* __restrict__ mask,
                                   int* __restrict__ counts, int rows) {
    int r = blockIdx.x * blockDim.x + threadIdx.x;
    if (r < rows) {
        int s = 0;
#pragma unroll 8
        for (int j = 0; j < PAD; ++j) s += (mask[r * PAD + j] != 0);
        counts[r] = s;
    }
}

// ---- Pass 1: per-block atom-count sums ----

__global__ void block_counts_kernel(const int* __restrict__ an,
                                    const int* __restrict__ counts,
                                    int* __restrict__ blockSums, int N) {
    __shared__ int red[APB];
    int t = threadIdx.x;
    int atom = blockIdx.x * APB + t;
    red[t] = (atom < N) ? counts[an[atom]] : 0;
    __syncthreads();
#pragma unroll
    for (int s = APB / 2; s > 0; s >>= 1) {
        if (t < s) red[t] += red[t + s];
        __syncthreads();
    }
    if (t == 0) blockSums[blockIdx.x] = red[0];
}

// ---- Pass 2: exclusive scan of block sums (single block, chunked) ----

__global__ void scan_blocksums_kernel(const int* __restrict__ sums,
                                      int* __restrict__ offs, int n) {
    __shared__ int buf[SCAN_TPB];
    __shared__ int carry;
    int t = threadIdx.x;
    if (t == 0) carry = 0;
    __syncthreads();
    for (int base = 0; base < n; base += SCAN_TPB) {
        int i = base + t;
        int v = (i < n) ? sums[i] : 0;
        buf[t] = v;
        __syncthreads();
        for (int d = 1; d < SCAN_TPB; d <<= 1) {
            int add = (t >= d) ? buf[t - d] : 0;
            __syncthreads();
            buf[t] += add;
            __syncthreads();
        }
        if (i < n) offs[i] = carry + buf[t] - v;   // exclusive
        int total = buf[SCAN_TPB - 1];
        __syncthreads();
        if (t == 0) carry += total;
        __syncthreads();
    }
}

// ---- Pass 3: intra-block offset scan + async-staged permutation gather ----

__global__ void gather_kernel(const int* __restrict__ an,
                              const float* __restrict__ src,
                              const int* __restrict__ omap,
                              const int* __restrict__ counts,
                              const int* __restrict__ blockOffs,
                              float* __restrict__ out, int N) {
    __shared__ int   s_k[APB];
    __shared__ int   s_e[APB];
    __shared__ int   s_off[APB];
    __shared__ int   s_scan[APB];
    __shared__ float s_row[8][2][PAD];   // [wave][buffer][element] = 8 KB

    const int t         = threadIdx.x;
    const int wave      = t >> 5;
    const int lane      = t & 31;
    const int blockStart = blockIdx.x * APB;

    // Kick off the first async row DMA for this wave ASAP (overlaps the scan).
    const uint32_t ldsB0 = (uint32_t)(uintptr_t)&s_row[wave][0][0] + (uint32_t)(lane * 16);
    const uint32_t ldsB1 = (uint32_t)(uintptr_t)&s_row[wave][1][0] + (uint32_t)(lane * 16);
    const int a0 = wave;
    const bool have0 = (blockStart + a0) < N;
    if (have0)
        async_row_b128(src + (size_t)(blockStart + a0) * PAD + lane * 4, ldsB0);

    // Per-atom counts + exclusive scan -> output offsets.
    int atom = blockStart + t;
    int e = 0, k = 0;
    if (atom < N) { e = an[atom]; k = counts[e]; }
    s_k[t] = k; s_e[t] = e; s_scan[t] = k;
    __syncthreads();
#pragma unroll
    for (int d = 1; d < APB; d <<= 1) {
        int add = (t >= d) ? s_scan[t - d] : 0;
        __syncthreads();
        s_scan[t] += add;
        __syncthreads();
    }
    s_off[t] = blockOffs[blockIdx.x] + s_scan[t] - k;
    __syncthreads();

    // Each wave consumes atoms a0, a0+8, ... with double-buffered async DMA.
    int parity = 0;
    for (int a = a0; a < APB; a += 8) {
        if (blockStart + a >= N) break;            // wave-uniform
        const int nxt = a + 8;
        const bool more = (nxt < APB) && (blockStart + nxt < N);
        if (more) {
            async_row_b128(src + (size_t)(blockStart + nxt) * PAD + lane * 4,
                           parity ? ldsB0 : ldsB1);
            wait_async_le1();                      // oldest DMA (current buf) done
        } else {
            wait_async_le0();                      // drain
        }

        const int e_a = s_e[a];
        const int off = s_off[a];
        const int k_a = s_k[a];
        const float* row = &s_row[wave][parity][0];
#pragma unroll
        for (int it = 0; it < 4; ++it) {
            int j = lane + 32 * it;
            int idx = omap[e_a * PAD + j];         // L2-hot, coalesced
            float val = row[idx];                  // LDS permutation gather
            if (j < k_a)
                __builtin_nontemporal_store(val, &out[off + j]);  // coalesced
        }
        parity ^= 1;
    }
}

// ---------------------------------------------------------------------------

extern "C" void kernel_launch(void* const* d_in, const int* in_sizes, int n_in,
                              void* d_out, int out_size, void* d_ws, size_t ws_size,
                              hipStream_t stream) {
    const int*   an    = (const int*)d_in[0];     // atomic_numbers (int)
    const float* src   = (const float*)d_in[1];   // [N, 128] f32
    const int*   omap  = (const int*)d_in[2];     // [rows, 128] int
    const int*   omask = (const int*)d_in[3];     // [rows, 128] int (0/1)

    const int N    = in_sizes[0];
    const int rows = in_sizes[2] / PAD;
    const int numBlocks = (N + APB - 1) / APB;

    char* ws = (char*)d_ws;
    int* counts    = (int*)ws;                    // 128 ints (512 B)
    int* blockSums = (int*)(ws + 512);            // numBlocks ints
    int* blockOffs = blockSums + numBlocks;       // numBlocks ints

    elem_counts_kernel<<<(rows + 127) / 128, 128, 0, stream>>>(omask, counts, rows);
    block_counts_kernel<<<numBlocks, APB, 0, stream>>>(an, counts, blockSums, N);
    scan_blocksums_kernel<<<1, SCAN_TPB, 0, stream>>>(blockSums, blockOffs, numBlocks);
    gather_kernel<<<numBlocks, GATHER_TPB, 0, stream>>>(an, src, omap, counts,
                                                        blockOffs, (float*)d_out, N);
}